// PolyMatchingLoss_13434657702249
// MI455X (gfx1250) — compile-verified
//
#include <hip/hip_runtime.h>

typedef __attribute__((ext_vector_type(2))) float v2f;
typedef __attribute__((ext_vector_type(8))) float v8f;

#define PNUM 128
#define BATCH 8192
#define WAVES_PER_BLOCK 8
#define NBLOCKS 512
#define TASKS_PER_WAVE 4   // 512 * 8 * 4 == 16384 == 2 * BATCH

// Async DMA: 16B per lane, global (SGPR base + VGPR off + IMM) -> LDS (VGPR off + IMM).
// IOFFSET is added to BOTH the LDS and global address (ISA 15.18.3 op 98).
#define ASYNC_LD16(ldsoff, voff, base, IMM)                                    \
    asm volatile("global_load_async_to_lds_b128 %0, %1, %2 offset:" #IMM       \
                 :: "v"(ldsoff), "v"(voff), "s"(base) : "memory")

#define ASYNC_WAIT(N) asm volatile("s_wait_asynccnt " #N ::: "memory")

__device__ __forceinline__ float sl1(float d) {
    float a = __builtin_fabsf(d);
    return a < 1.0f ? 0.5f * d * d : a - 0.5f;
}

// Stage one task: pred row (1KB), gt row (1KB) + duplicated gt copy (modulo-free).
__device__ __forceinline__ void stage_task(unsigned p_off, unsigned g_off,
                                           const float* pbase, const float* gbase,
                                           unsigned voff) {
    ASYNC_LD16(p_off,          voff, pbase, 0);
    ASYNC_LD16(p_off,          voff, pbase, 512);
    ASYNC_LD16(g_off,          voff, gbase, 0);
    ASYNC_LD16(g_off,          voff, gbase, 512);
    ASYNC_LD16(g_off + 1024u,  voff, gbase, 0);    // duplicate copy at +1KB
    ASYNC_LD16(g_off + 1024u,  voff, gbase, 512);
}

__global__ __launch_bounds__(256) void poly_match_kernel(
    const float* __restrict__ pred0,
    const float* __restrict__ pred1,
    const float* __restrict__ gt,
    float* __restrict__ partial)
{
    // Double-buffered per-wave staging: gt duplicated so (r+j)%128 is a flat offset.
    __shared__ __align__(16) float lds_gt[2][WAVES_PER_BLOCK][512];
    __shared__ __align__(16) float lds_pred[2][WAVES_PER_BLOCK][256];
    __shared__ float wave_out[WAVES_PER_BLOCK];

    const int tid  = threadIdx.x;
    const int wave = tid >> 5;
    const int lane = tid & 31;
    const unsigned voff = (unsigned)lane * 16u;

    float* gbufp[2];
    float* pbufp[2];
    unsigned g_off[2], p_off[2];
    #pragma unroll
    for (int k = 0; k < 2; ++k) {
        gbufp[k] = &lds_gt[k][wave][0];
        pbufp[k] = &lds_pred[k][wave][0];
        g_off[k] = (unsigned)(uintptr_t)gbufp[k] + voff;  // low 32 bits = LDS offset
        p_off[k] = (unsigned)(uintptr_t)pbufp[k] + voff;
    }

    const int wgid = blockIdx.x * WAVES_PER_BLOCK + wave;

    // WMMA 16x16x4 f32 A-fragment ownership (ISA 7.12.2):
    // lanes 0-15: M=lane,    K={0,1}; lanes 16-31: M=lane-16, K={2,3}
    const int M     = lane & 15;
    const int kbase = (lane >> 4) * 2;

    v2f ones;
    ones.x = 1.0f;
    ones.y = 1.0f;

    // Prologue: kick off task 0 staging.
    {
        const int task = wgid * TASKS_PER_WAVE;
        const int b = task >> 1;
        const float* pr = (task & 1) ? pred1 : pred0;
        stage_task(p_off[0], g_off[0], pr + (size_t)b * 256, gt + (size_t)b * 256, voff);
    }

    float acc = 0.0f;

    for (int t = 0; t < TASKS_PER_WAVE; ++t) {
        const int cur = t & 1;

        if (t + 1 < TASKS_PER_WAVE) {
            // Issue next task's 6 async loads into the other buffer, then wait
            // until only those 6 remain (async loads complete in order).
            const int ntask = wgid * TASKS_PER_WAVE + t + 1;
            const int nb = ntask >> 1;
            const float* npr = (ntask & 1) ? pred1 : pred0;
            stage_task(p_off[cur ^ 1], g_off[cur ^ 1],
                       npr + (size_t)nb * 256, gt + (size_t)nb * 256, voff);
            ASYNC_WAIT(6);
        } else {
            ASYNC_WAIT(0);
        }

        const float* pbuf = pbufp[cur];
        const float* gbuf = gbufp[cur];

        float task_min = 3.402823466e38f;

        // 128 rotations = 8 tiles of 16 (M dim); 128 j = 32 chunks of 4 (K dim).
        for (int tile = 0; tile < 8; ++tile) {
            const int r = tile * 16 + M;
            v8f c = {};  // D accumulates sum_j S[r, j] in every column

            // Software pipeline: prefetch chunk q+1 from LDS while computing q,
            // so s_wait_dscnt lands a full chunk of VALU after the ds_loads.
            float4 pv = *(const float4*)(pbuf + 2 * kbase);
            float2 g0 = *(const float2*)(gbuf + 2 * (r + kbase));
            float2 g1 = *(const float2*)(gbuf + 2 * (r + kbase) + 2);

            #pragma unroll 8
            for (int q = 0; q < 32; ++q) {
                const float4 pv_c = pv;
                const float2 g0_c = g0;
                const float2 g1_c = g1;

                // Prefetch next chunk (wraps to chunk 0 on last iter; dead value,
                // in-range, keeps the unrolled body branch-free).
                const int jn = ((q + 1) & 31) * 4 + kbase;
                pv = *(const float4*)(pbuf + 2 * jn);
                g0 = *(const float2*)(gbuf + 2 * (r + jn));
                g1 = *(const float2*)(gbuf + 2 * (r + jn) + 2);

                v2f a;
                a.x = sl1(pv_c.x - g0_c.x) + sl1(pv_c.y - g0_c.y);  // S[r, j0]
                a.y = sl1(pv_c.z - g1_c.x) + sl1(pv_c.w - g1_c.y);  // S[r, j0+1]
                c = __builtin_amdgcn_wmma_f32_16x16x4_f32(
                        /*neg_a=*/false, a, /*neg_b=*/false, ones,
                        /*c_mod=*/(short)0, c, /*reuse_a=*/false, /*reuse_b=*/false);
            }
            // Lane holds row sums for M = v + 8*(lane>=16); min over them.
            float m = c[0];
            m = fminf(m, c[1]); m = fminf(m, c[2]); m = fminf(m, c[3]);
            m = fminf(m, c[4]); m = fminf(m, c[5]); m = fminf(m, c[6]);
            m = fminf(m, c[7]);
            task_min = fminf(task_min, m);
        }
        // Halves are internally uniform; fold M 0-7 with M 8-15.
        task_min = fminf(task_min, __shfl_xor(task_min, 16, 32));

        // dis includes mean over j (1/128); final loss averages 16384 task mins.
        acc += task_min * (1.0f / (128.0f * (float)BATCH * 2.0f));
    }

    if (lane == 0) wave_out[wave] = acc;
    __syncthreads();
    if (tid == 0) {
        float s = 0.0f;
        #pragma unroll
        for (int w = 0; w < WAVES_PER_BLOCK; ++w) s += wave_out[w];
        partial[blockIdx.x] = s;
    }
}

__global__ __launch_bounds__(256) void poly_match_reduce(
    const float* __restrict__ partial, float* __restrict__ out)
{
    __shared__ float s[256];
    const int t = threadIdx.x;
    float v = 0.0f;
    for (int i = t; i < NBLOCKS; i += 256) v += partial[i];
    s[t] = v;
    __syncthreads();
    for (int stride = 128; stride > 0; stride >>= 1) {
        if (t < stride) s[t] += s[t + stride];
        __syncthreads();
    }
    if (t == 0) out[0] = s[0];
}

extern "C" void kernel_launch(void* const* d_in, const int* in_sizes, int n_in,
                              void* d_out, int out_size, void* d_ws, size_t ws_size,
                              hipStream_t stream) {
    const float* pred0 = (const float*)d_in[0];
    const float* pred1 = (const float*)d_in[1];
    const float* gtp   = (const float*)d_in[2];
    float* out     = (float*)d_out;
    float* partial = (float*)d_ws;   // NBLOCKS floats = 2 KB

    poly_match_kernel<<<NBLOCKS, 256, 0, stream>>>(pred0, pred1, gtp, partial);
    poly_match_reduce<<<1, 256, 0, stream>>>(partial, out);
}